// GNN2_caiyang_54949811585067
// MI455X (gfx1250) — compile-verified
//
#include <hip/hip_runtime.h>
#include <hip/hip_bf16.h>
#include <stdint.h>

// ---------------------------------------------------------------------------
// LightGCN sampling kernel for MI455X (gfx1250, wave32).
//
// Roofline: zero dense-matmul FLOPs (SpMM gather/scatter + index gathers), so
// no WMMA path applies. Optimization levers instead:
//  * 150000-bit "needed row" bitmask -> skip ~92% of the 4M edges entirely.
//  * wave32 ballot compaction of surviving edges.
//  * gfx1250 async global->LDS staging of col/val streams (ASYNCcnt).
//  * global_prefetch_b8 on the row stream.
//  * hardware global_atomic_add_f32 (no-return) for the segment sum.
// Embedding table (38.4MB), bitmask (19KB) and accumulator (38.4MB) all fit
// in the 192MB L2, so the random gathers/atomics stay on-chip.
// ---------------------------------------------------------------------------

#define EMB_D 64
#define SPMM_BLOCK 256
#define SPMM_WAVES (SPMM_BLOCK / 32)

// ---- zero accumulator + bitmask -------------------------------------------
__global__ void gnn_zero_kernel(float4* __restrict__ acc4, long accN4,
                                unsigned* __restrict__ mask, int maskW) {
  long t = (long)blockIdx.x * blockDim.x + threadIdx.x;
  if (t < accN4) acc4[t] = make_float4(0.f, 0.f, 0.f, 0.f);
  if (t < maskW) mask[t] = 0u;
}

// ---- mark rows that will actually be read out -----------------------------
__global__ void gnn_mark_kernel(const int* __restrict__ users,
                                const int* __restrict__ pos,
                                const int* __restrict__ neg,
                                int B, int NEGN, int n_user,
                                unsigned* __restrict__ mask) {
  int i = blockIdx.x * blockDim.x + threadIdx.x;
  int total = 2 * B + NEGN;
  if (i >= total) return;
  int r;
  if (i < B)           r = users[i];
  else if (i < 2 * B)  r = n_user + pos[i - B];
  else                 r = n_user + neg[i - 2 * B];
  atomicOr(&mask[((unsigned)r) >> 5], 1u << (r & 31));
}

// ---- filtered SpMM: acc[row] += val * e0[col] for masked rows -------------
__global__ __launch_bounds__(SPMM_BLOCK)
void gnn_spmm_kernel(const int* __restrict__ adj_row,
                     const int* __restrict__ adj_col,
                     const float* __restrict__ adj_val,
                     const float* __restrict__ uemb,
                     const float* __restrict__ iemb,
                     const unsigned* __restrict__ mask,
                     float* __restrict__ acc,
                     int nnz, int n_user) {
  __shared__ unsigned s_col[SPMM_WAVES * 32];
  __shared__ float    s_val[SPMM_WAVES * 32];

  const int lane = threadIdx.x & 31;
  const int wv   = threadIdx.x >> 5;
  // low 32 bits of a flat pointer to LDS == workgroup-relative LDS byte offset
  const unsigned colOff = (unsigned)(uintptr_t)(&s_col[(wv << 5) | lane]);
  const unsigned valOff = (unsigned)(uintptr_t)(&s_val[(wv << 5) | lane]);

  const long waveGid     = (long)blockIdx.x * SPMM_WAVES + wv;
  const long chunkStride = (long)gridDim.x * SPMM_WAVES * 32;

  for (long base = waveGid * 32; base < nnz; base += chunkStride) {
    const int e  = (int)base + lane;
    const int ec = (e < nnz) ? e : (nnz - 1);  // clamp: keep EXEC full

    // Stage col/val of this 32-edge chunk into LDS asynchronously while we
    // test the row stream against the bitmask. Tracked by ASYNCcnt.
    asm volatile("global_load_async_to_lds_b32 %0, %1, off"
                 :: "v"(colOff), "v"(adj_col + ec) : "memory");
    asm volatile("global_load_async_to_lds_b32 %0, %1, off"
                 :: "v"(valOff), "v"(adj_val + ec) : "memory");

    // Prefetch the next strided chunk of the row stream (global_prefetch_b8).
    {
      long pn = base + chunkStride;
      __builtin_prefetch(adj_row + ((pn < nnz) ? pn : base), 0, 1);
    }

    const int r = adj_row[ec];
    const bool hit =
        (e < nnz) && ((mask[((unsigned)r) >> 5] >> (r & 31)) & 1u);
    unsigned m = __builtin_amdgcn_ballot_w32(hit);

    // Always drain asyncs before LDS reuse next iteration (async LDS writes
    // are not ordered with each other).
    asm volatile("s_wait_asynccnt 0x0" ::: "memory");

    while (m) {
      const int j = __builtin_ctz(m);
      m &= m - 1;
      const int      rj = __builtin_amdgcn_readlane(r, j);
      const unsigned c  = s_col[(wv << 5) | j];   // LDS broadcast read
      const float    v  = s_val[(wv << 5) | j];
      const float* src = (c < (unsigned)n_user)
                             ? (uemb + (size_t)c * EMB_D)
                             : (iemb + (size_t)(c - n_user) * EMB_D);
      const float2 x = *(const float2*)(src + 2 * lane);  // 64 dims / wave
      float* d = acc + (size_t)rj * EMB_D + 2 * lane;
      const float p0 = v * x.x;
      const float p1 = v * x.y;
      asm volatile("global_atomic_add_f32 %0, %1, off"
                   :: "v"(d),     "v"(p0) : "memory");
      asm volatile("global_atomic_add_f32 %0, %1, off"
                   :: "v"(d + 1), "v"(p1) : "memory");
    }
  }
}

// ---- epilogue: out = (e0 + 3*acc)/4 gathered at users/pos/neg -------------
__global__ void gnn_out_kernel(const int* __restrict__ users,
                               const int* __restrict__ pos,
                               const int* __restrict__ neg,
                               const float* __restrict__ uemb,
                               const float* __restrict__ iemb,
                               const float* __restrict__ acc,
                               float* __restrict__ out,
                               int B, int NEGN, int n_user) {
  const int t = blockIdx.x * blockDim.x + threadIdx.x;  // one float2 each
  const int totalRows = 2 * B + NEGN;
  if (t >= totalRows * 32) return;
  const int row = t >> 5;
  const int k   = (t & 31) * 2;
  int r;
  if (row < B)           r = users[row];
  else if (row < 2 * B)  r = n_user + pos[row - B];
  else                   r = n_user + neg[row - 2 * B];
  const float* src = (r < n_user) ? (uemb + (size_t)r * EMB_D)
                                  : (iemb + (size_t)(r - n_user) * EMB_D);
  const float2 e = *(const float2*)(src + k);
  const float2 a = *(const float2*)(acc + (size_t)r * EMB_D + k);
  float2 o;
  o.x = 0.25f * (e.x + 3.f * a.x);
  o.y = 0.25f * (e.y + 3.f * a.y);
  *(float2*)(out + (size_t)row * EMB_D + k) = o;
}

// ---------------------------------------------------------------------------
extern "C" void kernel_launch(void* const* d_in, const int* in_sizes, int n_in,
                              void* d_out, int out_size, void* d_ws, size_t ws_size,
                              hipStream_t stream) {
  const int*   users = (const int*)d_in[0];
  const int*   pos   = (const int*)d_in[1];
  const int*   neg   = (const int*)d_in[2];
  // d_in[3] (mask) and d_in[4] (norm_adj) are unused placeholders.
  const float* uemb  = (const float*)d_in[5];
  const float* iemb  = (const float*)d_in[6];
  const int*   arow  = (const int*)d_in[7];
  const int*   acol  = (const int*)d_in[8];
  const float* aval  = (const float*)d_in[9];
  float*       out   = (float*)d_out;

  const int B      = in_sizes[0];
  const int NEGN   = in_sizes[2];
  const int n_user = in_sizes[5] / EMB_D;
  const int n_item = in_sizes[6] / EMB_D;
  const int N      = n_user + n_item;
  const int nnz    = in_sizes[7];

  // Workspace layout: [bitmask | accumulator]
  const int    maskW  = (N + 31) / 32;
  char*        ws     = (char*)d_ws;
  unsigned*    mask   = (unsigned*)ws;
  const size_t accOff = ((size_t)maskW * 4 + 255) & ~(size_t)255;
  float*       acc    = (float*)(ws + accOff);
  const long   accN4  = (long)N * EMB_D / 4;

  // 1) zero accumulator + bitmask (must happen every call: deterministic)
  {
    long threads = accN4;  // accN4 > maskW always here
    int  blocks  = (int)((threads + 255) / 256);
    gnn_zero_kernel<<<blocks, 256, 0, stream>>>((float4*)acc, accN4, mask, maskW);
  }
  // 2) mark needed rows
  {
    int total  = 2 * B + NEGN;
    int blocks = (total + 255) / 256;
    gnn_mark_kernel<<<blocks, 256, 0, stream>>>(users, pos, neg, B, NEGN, n_user, mask);
  }
  // 3) filtered SpMM with atomics
  {
    gnn_spmm_kernel<<<1024, SPMM_BLOCK, 0, stream>>>(
        arow, acol, aval, uemb, iemb, mask, acc, nnz, n_user);
  }
  // 4) gather epilogue
  {
    int total  = (2 * B + NEGN) * 32;
    int blocks = (total + 255) / 256;
    gnn_out_kernel<<<blocks, 256, 0, stream>>>(
        users, pos, neg, uemb, iemb, acc, out, B, NEGN, n_user);
  }
}